// SupConLoss_17995912970299
// MI455X (gfx1250) — compile-verified
//
#include <hip/hip_runtime.h>
#include <hip/hip_bf16.h>

typedef float v2f __attribute__((ext_vector_type(2)));
typedef float v4f __attribute__((ext_vector_type(4)));
typedef float v8f __attribute__((ext_vector_type(8)));

#define B_N 4096
#define D_K 128
#define NPAIR (D_K / 2)   // 64 K-pairs per row
#define EPSF 1e-8f

// ---------------- Kernel 1: packed-feature precompute -----------------
// Apack[row][p][8] = {var(2p),var(2p+1), mu,mu, e^-v,e^-v, mu*e^-v,mu*e^-v}
// Bpack[row][p][8] = {1/v,1/v, mu,mu, e^v,e^v, mu/v,mu/v}
// Row scalars: detv = prod(var) (natural fp32 underflow, matches reference),
//              svar = sum(var), rsum = sum(mu^2 e^-v), csum = sum(mu^2 / v)
__global__ __launch_bounds__(128)
void precompute_kernel(const float* __restrict__ mu, const float* __restrict__ var,
                       float* __restrict__ Apack, float* __restrict__ Bpack,
                       float* __restrict__ detv, float* __restrict__ svar,
                       float* __restrict__ rsum, float* __restrict__ csum)
{
    __shared__ float sprod[128], ssum[128], sr[128], sc[128];
    const int row = blockIdx.x;
    const int k   = threadIdx.x;
    const int idx = row * D_K + k;
    const float v = var[idx];
    const float m = mu[idx];
    const float iv = 1.0f / v;
    const float en = __expf(-v);
    const float ep = __expf(v);

    const int p    = k >> 1;
    const int s    = k & 1;
    const int base = (row * NPAIR + p) * 8;
    Apack[base + 0 + s] = v;
    Apack[base + 2 + s] = m;
    Apack[base + 4 + s] = en;
    Apack[base + 6 + s] = m * en;
    Bpack[base + 0 + s] = iv;
    Bpack[base + 2 + s] = m;
    Bpack[base + 4 + s] = ep;
    Bpack[base + 6 + s] = m * iv;

    const float m2 = m * m;
    sprod[k] = v; ssum[k] = v; sr[k] = m2 * en; sc[k] = m2 * iv;
    __syncthreads();
    for (int st = 64; st > 0; st >>= 1) {
        if (k < st) {
            sprod[k] *= sprod[k + st];
            ssum[k]  += ssum[k + st];
            sr[k]    += sr[k + st];
            sc[k]    += sc[k + st];
        }
        __syncthreads();
    }
    if (k == 0) {
        detv[row] = sprod[0]; svar[row] = ssum[0];
        rsum[row] = sr[0];    csum[row] = sc[0];
    }
}

// ---------------- Kernel 2: fused pairwise WMMA + epilogue -------------
// 256 threads = 8 waves (4 M x 2 N). Wave tile 16(M) x 32(N): two 16x16
// WMMA tiles sharing A fragments. Block tile 64x64. Per K-step:
// 6x global_load_b128 + 3x v_pk_mul_f32 + 12x v_wmma_f32_16x16x4_f32.
__global__ __launch_bounds__(256)
void pairwise_kernel(const int* __restrict__ labels,
                     const float* __restrict__ Apack, const float* __restrict__ Bpack,
                     const float* __restrict__ detv, const float* __restrict__ svar,
                     const float* __restrict__ rsum, const float* __restrict__ csum,
                     float* __restrict__ partials)
{
    const int tid  = threadIdx.x;
    const int wave = tid >> 5;
    const int lane = tid & 31;
    const int hl   = lane >> 4;     // selects K pair within fragment
    const int lm   = lane & 15;     // M (A side) / N (B side) index
    const int wm   = wave & 3;      // 4 waves along M
    const int wn   = wave >> 2;     // 2 waves along N
    const int rowBase = blockIdx.y * 64 + wm * 16;
    const int colBase = blockIdx.x * 64 + wn * 32;

    // pair index used at k-step ks is ks/4 + hl -> float offset 2*ks within row pack
    const float* aP  = Apack + ((size_t)(rowBase + lm)      * NPAIR + hl) * 8;
    const float* bP0 = Bpack + ((size_t)(colBase + lm)      * NPAIR + hl) * 8;
    const float* bP1 = Bpack + ((size_t)(colBase + 16 + lm) * NPAIR + hl) * 8;

    v8f tr0 = {}, qA0 = {}, cr0 = {}, ex0 = {}, q2a0 = {}, q2b0 = {};
    v8f tr1 = {}, qA1 = {}, cr1 = {}, ex1 = {}, q2a1 = {}, q2b1 = {};

    for (int ks = 0; ks < D_K; ks += 4) {
        const v4f A0 = *(const v4f*)(aP  + 2 * ks);      // var, mu
        const v4f A1 = *(const v4f*)(aP  + 2 * ks + 4);  // e^-v, mu*e^-v
        const v4f B0 = *(const v4f*)(bP0 + 2 * ks);      // 1/v, mu
        const v4f B1 = *(const v4f*)(bP0 + 2 * ks + 4);  // e^v, mu/v
        const v4f C0 = *(const v4f*)(bP1 + 2 * ks);
        const v4f C1 = *(const v4f*)(bP1 + 2 * ks + 4);

        const v2f va  = __builtin_shufflevector(A0, A0, 0, 1);
        const v2f ma  = __builtin_shufflevector(A0, A0, 2, 3);
        const v2f ea  = __builtin_shufflevector(A1, A1, 0, 1);
        const v2f mea = __builtin_shufflevector(A1, A1, 2, 3);
        const v2f m2a = ma * ma;

        const v2f ib0  = __builtin_shufflevector(B0, B0, 0, 1);
        const v2f mb0  = __builtin_shufflevector(B0, B0, 2, 3);
        const v2f pb0  = __builtin_shufflevector(B1, B1, 0, 1);
        const v2f mib0 = __builtin_shufflevector(B1, B1, 2, 3);
        const v2f m2b0 = mb0 * mb0;

        const v2f ib1  = __builtin_shufflevector(C0, C0, 0, 1);
        const v2f mb1  = __builtin_shufflevector(C0, C0, 2, 3);
        const v2f pb1  = __builtin_shufflevector(C1, C1, 0, 1);
        const v2f mib1 = __builtin_shufflevector(C1, C1, 2, 3);
        const v2f m2b1 = mb1 * mb1;

        tr0  = __builtin_amdgcn_wmma_f32_16x16x4_f32(false, va,  false, ib0,  (short)0, tr0,  false, false);
        qA0  = __builtin_amdgcn_wmma_f32_16x16x4_f32(false, m2a, false, ib0,  (short)0, qA0,  false, false);
        cr0  = __builtin_amdgcn_wmma_f32_16x16x4_f32(false, ma,  false, mib0, (short)0, cr0,  false, false);
        ex0  = __builtin_amdgcn_wmma_f32_16x16x4_f32(false, ea,  false, pb0,  (short)0, ex0,  false, false);
        q2a0 = __builtin_amdgcn_wmma_f32_16x16x4_f32(false, ea,  false, m2b0, (short)0, q2a0, false, false);
        q2b0 = __builtin_amdgcn_wmma_f32_16x16x4_f32(false, mea, false, mb0,  (short)0, q2b0, false, false);

        tr1  = __builtin_amdgcn_wmma_f32_16x16x4_f32(false, va,  false, ib1,  (short)0, tr1,  false, false);
        qA1  = __builtin_amdgcn_wmma_f32_16x16x4_f32(false, m2a, false, ib1,  (short)0, qA1,  false, false);
        cr1  = __builtin_amdgcn_wmma_f32_16x16x4_f32(false, ma,  false, mib1, (short)0, cr1,  false, false);
        ex1  = __builtin_amdgcn_wmma_f32_16x16x4_f32(false, ea,  false, pb1,  (short)0, ex1,  false, false);
        q2a1 = __builtin_amdgcn_wmma_f32_16x16x4_f32(false, ea,  false, m2b1, (short)0, q2a1, false, false);
        q2b1 = __builtin_amdgcn_wmma_f32_16x16x4_f32(false, mea, false, mb1,  (short)0, q2b1, false, false);
    }

    // Epilogue. C/D layout: element v of lane -> i = rowBase + v + 8*hl,
    // j = colBase + 16*tile + lm.
    const int   j0 = colBase + lm;
    const int   j1 = colBase + 16 + lm;
    const float detv_j0 = detv[j0], svar_j0 = svar[j0], csum_j0 = csum[j0];
    const float detv_j1 = detv[j1], svar_j1 = svar[j1], csum_j1 = csum[j1];
    const int   lab_j0 = labels[j0], lab_j1 = labels[j1];

    float pos = 0.0f, neg = 0.0f;
#pragma unroll
    for (int v = 0; v < 8; ++v) {
        const int   i      = rowBase + v + 8 * hl;
        const float detv_i = detv[i];
        const float svar_i = svar[i];
        const float rsum_i = rsum[i];
        const int   lab_i  = labels[i];

        {   // tile 0
            const float kl_a = 0.5f * (tr0[v] + detv_j0 / (detv_i + EPSF)
                                       + (qA0[v] - 2.0f * cr0[v] + csum_j0) - (float)D_K);
            const float kl_b = 0.5f * (-(float)D_K + (svar_i - svar_j0) + ex0[v]
                                       + (q2a0[v] - 2.0f * q2b0[v] + rsum_i));
            const float sym = 0.5f * (kl_a + kl_b);
            const float lg  = 1.0f / (1.0f + __expf(-sym));
            if (i != j0) { if (lab_i == lab_j0) pos += lg; else neg += lg; }
        }
        {   // tile 1
            const float kl_a = 0.5f * (tr1[v] + detv_j1 / (detv_i + EPSF)
                                       + (qA1[v] - 2.0f * cr1[v] + csum_j1) - (float)D_K);
            const float kl_b = 0.5f * (-(float)D_K + (svar_i - svar_j1) + ex1[v]
                                       + (q2a1[v] - 2.0f * q2b1[v] + rsum_i));
            const float sym = 0.5f * (kl_a + kl_b);
            const float lg  = 1.0f / (1.0f + __expf(-sym));
            if (i != j1) { if (lab_i == lab_j1) pos += lg; else neg += lg; }
        }
    }

    // wave32 reduction, then per-block partial (no fp atomics -> deterministic)
    for (int off = 16; off > 0; off >>= 1) {
        pos += __shfl_xor(pos, off, 32);
        neg += __shfl_xor(neg, off, 32);
    }
    __shared__ float sp[8], sn[8];
    if (lane == 0) { sp[wave] = pos; sn[wave] = neg; }
    __syncthreads();
    if (tid == 0) {
        float tp = 0.0f, tn = 0.0f;
        for (int w = 0; w < 8; ++w) { tp += sp[w]; tn += sn[w]; }
        const int bid = blockIdx.y * gridDim.x + blockIdx.x;
        partials[2 * bid + 0] = tp;
        partials[2 * bid + 1] = tn;
    }
}

// ---------------- Kernel 3: deterministic final reduction --------------
__global__ __launch_bounds__(256)
void reduce_kernel(const float* __restrict__ partials, float* __restrict__ out, int npart)
{
    __shared__ float sp[256], sn[256];
    const int tid = threadIdx.x;
    float p = 0.0f, n = 0.0f;
    for (int i = tid; i < npart; i += 256) {
        p += partials[2 * i + 0];
        n += partials[2 * i + 1];
    }
    sp[tid] = p; sn[tid] = n;
    __syncthreads();
    for (int s = 128; s > 0; s >>= 1) {
        if (tid < s) { sp[tid] += sp[tid + s]; sn[tid] += sn[tid + s]; }
        __syncthreads();
    }
    if (tid == 0) {
        const float inv = 1.0f / ((float)B_N * (float)B_N);
        out[0] = sp[0] * inv;   // mean(logits * p_mask)
        out[1] = sn[0] * inv;   // mean(logits * n_mask)
    }
}

extern "C" void kernel_launch(void* const* d_in, const int* in_sizes, int n_in,
                              void* d_out, int out_size, void* d_ws, size_t ws_size,
                              hipStream_t stream) {
    const float* mu     = (const float*)d_in[0];
    const float* var    = (const float*)d_in[1];
    const int*   labels = (const int*)d_in[2];

    float* ws    = (float*)d_ws;
    float* Apack = ws;                             // 4096 * 64 * 8 = 2M floats
    float* Bpack = Apack + (size_t)B_N * NPAIR * 8;
    float* detv  = Bpack + (size_t)B_N * NPAIR * 8;
    float* svar  = detv + B_N;
    float* rsum  = svar + B_N;
    float* csum  = rsum + B_N;
    float* partials = csum + B_N;                  // 2 * (64*64)

    precompute_kernel<<<dim3(B_N), dim3(D_K), 0, stream>>>(
        mu, var, Apack, Bpack, detv, svar, rsum, csum);

    pairwise_kernel<<<dim3(B_N / 64, B_N / 64), dim3(256), 0, stream>>>(
        labels, Apack, Bpack, detv, svar, rsum, csum, partials);

    reduce_kernel<<<dim3(1), dim3(256), 0, stream>>>(
        partials, (float*)d_out, (B_N / 64) * (B_N / 64));
}